// DirectedGCN_52201032515658
// MI455X (gfx1250) — compile-verified
//
#include <hip/hip_runtime.h>

#define D_FEAT 128

typedef float v2f __attribute__((ext_vector_type(2)));
typedef float v8f __attribute__((ext_vector_type(8)));

// ---------------------------------------------------------------------------
// Degree accumulation: deg[col] += ew[e]   (deg buffer pre-zeroed)
// ---------------------------------------------------------------------------
__global__ __launch_bounds__(256) void gcn_deg_kernel(
    const long long* __restrict__ ei, const float* __restrict__ ew,
    float* __restrict__ deg, int nE)
{
    int e = blockIdx.x * 256 + threadIdx.x;
    if (e < nE) {
        long long col = ei[(size_t)nE + e];
        atomicAdd(deg + col, ew[e]);
    }
}

// ---------------------------------------------------------------------------
// dinv[n] = rsqrt(deg[n] + 1)   (in place; deg >= 0 so deg+1 > 0 always)
// ---------------------------------------------------------------------------
__global__ __launch_bounds__(256) void gcn_dinv_kernel(float* __restrict__ dinv, int nN)
{
    int i = blockIdx.x * 256 + threadIdx.x;
    if (i < nN) dinv[i] = rsqrtf(dinv[i] + 1.0f);
}

// ---------------------------------------------------------------------------
// Dense GEMM  Out[16-row stripe] = A[stripe] @ W   via V_WMMA_F32_16X16X4_F32
// Block = 256 threads = 8 waves; wave w computes the 16x16 tile at columns
// [16w, 16w+16). A-tile (16x128) staged in LDS with padded stride 132 floats.
// ---------------------------------------------------------------------------
__global__ __launch_bounds__(256) void gcn_gemm_wmma(
    const float* __restrict__ A, const float* __restrict__ W,
    float* __restrict__ Out)
{
    __shared__ float Alds[16 * 132];
    const int tid  = threadIdx.x;
    const int wave = tid >> 5;
    const int lane = tid & 31;
    const long long mbase = (long long)blockIdx.x * 16;

    // Cooperative, coalesced load of the 16x128 A tile (512 float4's).
    for (int f = tid; f < 512; f += 256) {
        const int r  = f >> 5;        // row 0..15
        const int c4 = f & 31;        // float4 column 0..31
        const float4 v = reinterpret_cast<const float4*>(A + (mbase + r) * D_FEAT)[c4];
        *reinterpret_cast<float4*>(&Alds[r * 132 + (c4 << 2)]) = v;
    }
    __syncthreads();

    // Fragment coordinates per the CDNA5 ISA f32 layouts (CDNA5 ISA):
    //   A 16x4 : lane l -> M = l&15, K = k0 + 2*(l>>4) + {0,1}
    //   B 4x16 : lane l -> N = l&15, K = k0 + 2*(l>>4) + {0,1}
    const int m    = lane & 15;
    const int koff = (lane >> 4) << 1;             // 0 or 2
    const int ncol = (wave << 4) + (lane & 15);

    v8f acc = {};
#pragma unroll
    for (int k0 = 0; k0 < D_FEAT; k0 += 4) {
        v2f a, b;
        a.x = Alds[m * 132 + k0 + koff];
        a.y = Alds[m * 132 + k0 + koff + 1];
        b.x = W[(k0 + koff)     * D_FEAT + ncol];
        b.y = W[(k0 + koff + 1) * D_FEAT + ncol];
        acc = __builtin_amdgcn_wmma_f32_16x16x4_f32(
            /*neg_a=*/false, a, /*neg_b=*/false, b,
            /*c_mod=*/(short)0, acc, /*reuse_a=*/false, /*reuse_b=*/false);
    }

    // C/D layout: VGPR r -> row (r + 8*(lane>>4)), col = lane&15.
    const int rowhi = (lane >> 4) << 3;            // 0 or 8
    const long long obase = (mbase + rowhi) * D_FEAT + (wave << 4) + (lane & 15);
#pragma unroll
    for (int r = 0; r < 8; ++r)
        Out[obase + (long long)r * D_FEAT] = acc[r];
}

// ---------------------------------------------------------------------------
// Edge aggregation: agg[col] += dinv[row]*ew*dinv[col] * h[row]
// One wave32 per edge; lane l owns float4 slice [4l, 4l+4) of the 128 dims.
// h/agg are L2-resident (51.2 MB each) so random traffic stays on-die.
// ---------------------------------------------------------------------------
__global__ __launch_bounds__(256) void gcn_edge_agg(
    const long long* __restrict__ ei, const float* __restrict__ ew,
    const float* __restrict__ dinv, const float* __restrict__ h,
    float* __restrict__ agg, int nE)
{
    const int e = blockIdx.x * 8 + (threadIdx.x >> 5);
    if (e >= nE) return;
    const int lane = threadIdx.x & 31;

    const long long row = ei[e];
    const long long col = ei[(size_t)nE + e];
    const float w = dinv[row] * ew[e] * dinv[col];

    const float4 hv = reinterpret_cast<const float4*>(h + row * D_FEAT)[lane];
    float* dst = agg + col * D_FEAT + (lane << 2);
    atomicAdd(dst + 0, w * hv.x);
    atomicAdd(dst + 1, w * hv.y);
    atomicAdd(dst + 2, w * hv.z);
    atomicAdd(dst + 3, w * hv.w);
}

// ---------------------------------------------------------------------------
// io = [relu]( io + dinv[node]^2 * h + bias )    (float4 per thread, in place)
// ---------------------------------------------------------------------------
__global__ __launch_bounds__(256) void gcn_combine(
    float* __restrict__ io, const float* __restrict__ h,
    const float* __restrict__ dinv, const float* __restrict__ bias,
    int nN, int doRelu)
{
    const int i4 = blockIdx.x * 256 + threadIdx.x;     // float4 index
    if (i4 >= nN * (D_FEAT / 4)) return;
    const int node = i4 >> 5;                           // 32 float4 per node
    const int c4   = (i4 & 31) << 2;

    const float d  = dinv[node];
    const float dd = d * d;
    const float4 a  = reinterpret_cast<const float4*>(io)[i4];
    const float4 hv = reinterpret_cast<const float4*>(h)[i4];
    const float4 bv = *reinterpret_cast<const float4*>(bias + c4);

    float4 r;
    r.x = a.x + dd * hv.x + bv.x;
    r.y = a.y + dd * hv.y + bv.y;
    r.z = a.z + dd * hv.z + bv.z;
    r.w = a.w + dd * hv.w + bv.w;
    if (doRelu) {
        r.x = fmaxf(r.x, 0.0f); r.y = fmaxf(r.y, 0.0f);
        r.z = fmaxf(r.z, 0.0f); r.w = fmaxf(r.w, 0.0f);
    }
    reinterpret_cast<float4*>(io)[i4] = r;
}

// ---------------------------------------------------------------------------
extern "C" void kernel_launch(void* const* d_in, const int* in_sizes, int n_in,
                              void* d_out, int out_size, void* d_ws, size_t ws_size,
                              hipStream_t stream)
{
    const float*     x  = (const float*)d_in[0];
    const long long* ei = (const long long*)d_in[1];   // int64 edge_index [2,E]
    const float*     ew = (const float*)d_in[2];
    const float*     W1 = (const float*)d_in[3];
    const float*     b1 = (const float*)d_in[4];
    const float*     W2 = (const float*)d_in[5];
    const float*     b2 = (const float*)d_in[6];
    float* out = (float*)d_out;

    const int nE = in_sizes[2];                // 1,600,000
    const int nN = in_sizes[0] / D_FEAT;       // 100,000 (divisible by 16)

    // Workspace layout: dinv | hbuf (N x 128) | aggbuf (N x 128)
    char* ws = (char*)d_ws;
    const size_t featBytes = (size_t)nN * D_FEAT * sizeof(float);
    size_t off = 0;
    float* dinv = (float*)(ws + off);
    off += ((size_t)nN * sizeof(float) + 255) & ~(size_t)255;
    float* hbuf = (float*)(ws + off);
    off += (featBytes + 255) & ~(size_t)255;
    float* aggbuf = (float*)(ws + off);

    // Accumulator buffers must be zero every call (harness does not re-zero).
    hipMemsetAsync(dinv,   0, (size_t)nN * sizeof(float), stream);
    hipMemsetAsync(aggbuf, 0, featBytes, stream);
    hipMemsetAsync(out,    0, featBytes, stream);

    const int edgeBlocks1 = (nE + 255) / 256;
    const int edgeBlocks8 = (nE + 7) / 8;       // 8 edges (waves) per block
    const int nodeBlocks  = (nN + 255) / 256;
    const int combBlocks  = (nN * (D_FEAT / 4) + 255) / 256;
    const int gemmBlocks  = nN / 16;

    // Shared normalization: deg = segsum(ew, col); dinv = rsqrt(deg + 1)
    gcn_deg_kernel <<<edgeBlocks1, 256, 0, stream>>>(ei, ew, dinv, nE);
    gcn_dinv_kernel<<<nodeBlocks,  256, 0, stream>>>(dinv, nN);

    // ---- Layer 1 ----
    gcn_gemm_wmma<<<gemmBlocks, 256, 0, stream>>>(x, W1, hbuf);
    gcn_edge_agg <<<edgeBlocks8, 256, 0, stream>>>(ei, ew, dinv, hbuf, aggbuf, nE);
    gcn_combine  <<<combBlocks, 256, 0, stream>>>(aggbuf, hbuf, dinv, b1, nN, 1); // -> hid

    // ---- Layer 2 ----
    gcn_gemm_wmma<<<gemmBlocks, 256, 0, stream>>>(aggbuf, W2, hbuf);
    gcn_edge_agg <<<edgeBlocks8, 256, 0, stream>>>(ei, ew, dinv, hbuf, out, nE);
    gcn_combine  <<<combBlocks, 256, 0, stream>>>(out, hbuf, dinv, b2, nN, 0);
}